// GNNActor_1752346657346
// MI455X (gfx1250) — compile-verified
//
#include <hip/hip_runtime.h>
#include <hip/hip_bf16.h>

#define N_NODES 98304
#define E_EDGES 1572864
#define C_DIM   64
#define H_DIM   32

typedef float v2f __attribute__((ext_vector_type(2)));
typedef float v8f __attribute__((ext_vector_type(8)));

// ---------------- degree / normalization ----------------

__global__ void __launch_bounds__(256) k_init_deg(float* __restrict__ deg) {
    int i = blockIdx.x * 256 + threadIdx.x;     // grid covers exactly N
    deg[i] = 1.0f;                               // self-loop
}

__global__ void __launch_bounds__(256) k_count_deg(const long long* __restrict__ ei,
                                                   float* __restrict__ deg) {
    int e = blockIdx.x * 256 + threadIdx.x;     // grid covers exactly E
    long long d = ei[(size_t)E_EDGES + e];      // dst row
    atomicAdd(&deg[d], 1.0f);
}

__global__ void __launch_bounds__(256) k_rsqrt_deg(float* __restrict__ deg) {
    int i = blockIdx.x * 256 + threadIdx.x;
    deg[i] = rsqrtf(deg[i]);
}

// ---------------- h = state @ conv_W  (WMMA f32 16x16x4), agg init ----------------
// 8 waves / block, each wave owns one 16-row strip and all 4 col tiles (64 cols).

__global__ void __launch_bounds__(256) k_gemm_conv(const float* __restrict__ state,
                                                   const float* __restrict__ W,
                                                   const float* __restrict__ dinv,
                                                   float* __restrict__ h,
                                                   float* __restrict__ agg) {
    __shared__ float sW[64 * 65];               // padded: bank-conflict-free B reads
    for (int i = threadIdx.x; i < 64 * 64; i += 256) {
        int r = i >> 6, c = i & 63;
        sW[r * 65 + c] = W[i];
    }
    __syncthreads();

    const int wave  = threadIdx.x >> 5;
    const int lane  = threadIdx.x & 31;
    const int lr    = lane & 15;
    const int hi    = lane >> 4;
    const int mtile = blockIdx.x * 8 + wave;    // 16-row strip
    const int rowb  = mtile * 16;

    v8f acc0 = {}, acc1 = {}, acc2 = {}, acc3 = {};
    const float* __restrict__ arow = state + (size_t)(rowb + lr) * C_DIM;

    #pragma unroll
    for (int k = 0; k < 64; k += 4) {
        const int kb = k + hi * 2;              // ISA A layout: lanes>=16 hold K+2,K+3
        v2f a; a.x = arow[kb]; a.y = arow[kb + 1];
        const float* w0 = &sW[kb * 65];
        const float* w1 = &sW[(kb + 1) * 65];
        v2f b0; b0.x = w0[ 0 + lr]; b0.y = w1[ 0 + lr];
        v2f b1; b1.x = w0[16 + lr]; b1.y = w1[16 + lr];
        v2f b2; b2.x = w0[32 + lr]; b2.y = w1[32 + lr];
        v2f b3; b3.x = w0[48 + lr]; b3.y = w1[48 + lr];
        acc0 = __builtin_amdgcn_wmma_f32_16x16x4_f32(false, a, false, b0, (short)0, acc0, false, false);
        acc1 = __builtin_amdgcn_wmma_f32_16x16x4_f32(false, a, false, b1, (short)0, acc1, false, false);
        acc2 = __builtin_amdgcn_wmma_f32_16x16x4_f32(false, a, false, b2, (short)0, acc2, false, false);
        acc3 = __builtin_amdgcn_wmma_f32_16x16x4_f32(false, a, false, b3, (short)0, acc3, false, false);
    }

    #pragma unroll
    for (int i = 0; i < 8; ++i) {               // D layout: vgpr i -> M=i (+8 for hi half)
        int row = rowb + hi * 8 + i;
        float dv  = dinv[row];
        float nrm = dv * dv;                    // self-loop norm
        float* hr = h   + (size_t)row * C_DIM + lr;
        float* ar = agg + (size_t)row * C_DIM + lr;
        hr[0]  = acc0[i];  ar[0]  = acc0[i] * nrm;
        hr[16] = acc1[i];  ar[16] = acc1[i] * nrm;
        hr[32] = acc2[i];  ar[32] = acc2[i] * nrm;
        hr[48] = acc3[i];  ar[48] = acc3[i] * nrm;
    }
}

// ---------------- edge scatter: agg[dst] += h[src] * dinv[src]*dinv[dst] ----------------

__global__ void __launch_bounds__(256) k_scatter(const long long* __restrict__ ei,
                                                 const float* __restrict__ h,
                                                 const float* __restrict__ dinv,
                                                 float* __restrict__ agg) {
    int t = blockIdx.x * 256 + threadIdx.x;     // grid covers exactly E*64
    int e = t >> 6;
    int c = t & 63;
    long long s = ei[e];
    long long d = ei[(size_t)E_EDGES + e];
    float nrm = dinv[s] * dinv[d];
    atomicAdd(&agg[(size_t)d * C_DIM + c], h[(size_t)s * C_DIM + c] * nrm);
}

// ---------------- fused epilogue + MLP (WMMA chains) ----------------
// 4 waves / block; each wave handles a 16-node tile end to end.

__global__ void __launch_bounds__(128) k_mlp(const float* __restrict__ state,
                                             const float* __restrict__ agg,
                                             const float* __restrict__ conv_b,
                                             const float* __restrict__ W1,
                                             const float* __restrict__ b1,
                                             const float* __restrict__ W2,
                                             const float* __restrict__ b2,
                                             const float* __restrict__ W3,
                                             const float* __restrict__ b3,
                                             float* __restrict__ out) {
    __shared__ float sW1[64 * 33];
    __shared__ float sW2[32 * 33];
    __shared__ float sW3[32];
    __shared__ float sB1[32];
    __shared__ float sB2[32];
    __shared__ float sCB[64];
    __shared__ float sX [4][16 * 68];           // stride 68: float4-aligned + conflict-free
    __shared__ float sY1[4][16 * 36];
    __shared__ float sY2[4][16 * 33];

    for (int i = threadIdx.x; i < 64 * 32; i += 128) sW1[(i >> 5) * 33 + (i & 31)] = W1[i];
    for (int i = threadIdx.x; i < 32 * 32; i += 128) sW2[(i >> 5) * 33 + (i & 31)] = W2[i];
    if (threadIdx.x < 32) {
        sW3[threadIdx.x] = W3[threadIdx.x];
        sB1[threadIdx.x] = b1[threadIdx.x];
        sB2[threadIdx.x] = b2[threadIdx.x];
    }
    if (threadIdx.x < 64) sCB[threadIdx.x] = conv_b[threadIdx.x];
    const float bb3 = b3[0];
    __syncthreads();

    const int wave  = threadIdx.x >> 5;
    const int lane  = threadIdx.x & 31;
    const int lr    = lane & 15;
    const int hi    = lane >> 4;
    const int tile  = blockIdx.x * 4 + wave;
    const int nodeb = tile * 16;

    float* xw  = &sX[wave][0];
    float* y1w = &sY1[wave][0];
    float* y2w = &sY2[wave][0];

    // stage x = relu(agg + conv_b) + state (16 x 64, float4 coalesced per wave)
    const float4* av4 = (const float4*)(agg   + (size_t)nodeb * C_DIM);
    const float4* sv4 = (const float4*)(state + (size_t)nodeb * C_DIM);
    #pragma unroll
    for (int it = 0; it < 8; ++it) {
        int idx = it * 32 + lane;               // 256 float4 slots
        int r = idx >> 4, c4 = idx & 15;
        float4 a = av4[idx];
        float4 s = sv4[idx];
        float4 x;
        x.x = fmaxf(a.x + sCB[c4 * 4 + 0], 0.f) + s.x;
        x.y = fmaxf(a.y + sCB[c4 * 4 + 1], 0.f) + s.y;
        x.z = fmaxf(a.z + sCB[c4 * 4 + 2], 0.f) + s.z;
        x.w = fmaxf(a.w + sCB[c4 * 4 + 3], 0.f) + s.w;
        *(float4*)&xw[r * 68 + c4 * 4] = x;
    }
    __syncthreads();

    // layer 1: [16x64] @ [64x32]
    v8f p0 = {}, p1 = {};
    #pragma unroll
    for (int k = 0; k < 64; k += 4) {
        const int kb = k + hi * 2;
        v2f a; a.x = xw[lr * 68 + kb]; a.y = xw[lr * 68 + kb + 1];
        const float* w0 = &sW1[kb * 33];
        const float* w1 = &sW1[(kb + 1) * 33];
        v2f bl; bl.x = w0[lr];      bl.y = w1[lr];
        v2f bh; bh.x = w0[16 + lr]; bh.y = w1[16 + lr];
        p0 = __builtin_amdgcn_wmma_f32_16x16x4_f32(false, a, false, bl, (short)0, p0, false, false);
        p1 = __builtin_amdgcn_wmma_f32_16x16x4_f32(false, a, false, bh, (short)0, p1, false, false);
    }
    #pragma unroll
    for (int i = 0; i < 8; ++i) {
        int row = hi * 8 + i;
        y1w[row * 36 + lr]      = fmaxf(p0[i] + sB1[lr],      0.f);
        y1w[row * 36 + 16 + lr] = fmaxf(p1[i] + sB1[16 + lr], 0.f);
    }
    __syncthreads();

    // layer 2: [16x32] @ [32x32]
    v8f q0 = {}, q1 = {};
    #pragma unroll
    for (int k = 0; k < 32; k += 4) {
        const int kb = k + hi * 2;
        v2f a; a.x = y1w[lr * 36 + kb]; a.y = y1w[lr * 36 + kb + 1];
        const float* w0 = &sW2[kb * 33];
        const float* w1 = &sW2[(kb + 1) * 33];
        v2f bl; bl.x = w0[lr];      bl.y = w1[lr];
        v2f bh; bh.x = w0[16 + lr]; bh.y = w1[16 + lr];
        q0 = __builtin_amdgcn_wmma_f32_16x16x4_f32(false, a, false, bl, (short)0, q0, false, false);
        q1 = __builtin_amdgcn_wmma_f32_16x16x4_f32(false, a, false, bh, (short)0, q1, false, false);
    }
    #pragma unroll
    for (int i = 0; i < 8; ++i) {
        int row = hi * 8 + i;
        y2w[row * 33 + lr]      = fmaxf(q0[i] + sB2[lr],      0.f);
        y2w[row * 33 + 16 + lr] = fmaxf(q1[i] + sB2[16 + lr], 0.f);
    }
    __syncthreads();

    // head: [16x32] . W3 + b3 -> softplus -> +1e-20
    if (lane < 16) {
        float s = 0.f;
        #pragma unroll
        for (int c = 0; c < 32; ++c) s += y2w[lane * 33 + c] * sW3[c];
        float t  = s + bb3;
        float sp = (t > 20.f) ? t : log1pf(expf(t));
        out[nodeb + lane] = sp + 1e-20f;
    }
}

// ---------------- launcher ----------------

extern "C" void kernel_launch(void* const* d_in, const int* in_sizes, int n_in,
                              void* d_out, int out_size, void* d_ws, size_t ws_size,
                              hipStream_t stream) {
    const float*     state  = (const float*)d_in[0];
    const long long* ei     = (const long long*)d_in[1];
    const float*     conv_W = (const float*)d_in[2];
    const float*     conv_b = (const float*)d_in[3];
    const float*     W1     = (const float*)d_in[4];
    const float*     b1     = (const float*)d_in[5];
    const float*     W2     = (const float*)d_in[6];
    const float*     b2     = (const float*)d_in[7];
    const float*     W3     = (const float*)d_in[8];
    const float*     b3     = (const float*)d_in[9];
    float* out = (float*)d_out;

    float* deg = (float*)d_ws;                         // N
    float* h   = deg + N_NODES;                        // N*64
    float* agg = h + (size_t)N_NODES * C_DIM;          // N*64

    k_init_deg <<<N_NODES / 256, 256, 0, stream>>>(deg);
    k_count_deg<<<E_EDGES / 256, 256, 0, stream>>>(ei, deg);
    k_rsqrt_deg<<<N_NODES / 256, 256, 0, stream>>>(deg);

    // 6144 row-tiles, 8 waves/block -> 768 blocks
    k_gemm_conv<<<(N_NODES / 16) / 8, 256, 0, stream>>>(state, conv_W, deg, h, agg);

    // E*64 threads / 256 -> E/4 blocks
    k_scatter<<<E_EDGES / 4, 256, 0, stream>>>(ei, h, deg, agg);

    // 6144 node-tiles, 4 waves/block -> 1536 blocks
    k_mlp<<<(N_NODES / 16) / 4, 128, 0, stream>>>(state, agg, conv_b,
                                                  W1, b1, W2, b2, W3, b3, out);
}